// PACE_87170656240148
// MI455X (gfx1250) — compile-verified
//
#include <hip/hip_runtime.h>
#include <math.h>

typedef float v2f __attribute__((ext_vector_type(2)));
typedef float v8f __attribute__((ext_vector_type(8)));

#define BATCH   8192
#define NPTS    128
#define KBASIS  64
#define TN      384        // 3*NPTS
#define LAM     1.0f

// ---- workspace layout (float offsets) ----
#define WS_SQW   0          // 128
#define WS_WSUM  128        // 1
#define WS_Q00   129        // 1
#define WS_BW    160        // 192  b_w[k*3+j]
#define WS_G0    352        // 64   g[k]
#define WS_HT    416        // 384  htop[m]
#define WS_MH    800        // 384  (M^T h)[m]
#define WS_BARB  1184       // 24576 bar_B[m*64+k]
#define WS_AMAT  25760      // 24576 A_mat[k*384+m]
#define WS_BARH  50336      // 4096  bar_H[k*64+k2]
#define WS_BHA   54432      // 24576 (bar_H@A)[k*384+m]
#define WS_S     79008      // 147456 S[m*384+m2]
#define WS_YW    226464     // 24576 y_w[b*3+j]
#define WS_BARY  251040     // 3145728 bar_y[b*384 + j*128 + n]
#define WS_QBUF  3396768    // 819200 Q[b*100 + r*10 + c]
#define WS_SPQ   4215968    // 147456 interleaved S_pq^T for coalesced B-frags

// ============================================================ prep1
__global__ __launch_bounds__(128) void pace_prep1(const float* __restrict__ w,
                                                  const float* __restrict__ bkp,
                                                  float* __restrict__ ws) {
  __shared__ float aug[64][130];
  __shared__ float fac[64];
  __shared__ float sh_sqw[128];
  __shared__ float sh_bw[192];
  __shared__ float sh_g[64];
  __shared__ float sh_ht[384];
  __shared__ float sh_u[64];
  __shared__ float red[128];
  __shared__ float sc[2];
  const int t = threadIdx.x;

  float wv = w[t];
  red[t] = wv;
  sh_sqw[t] = sqrtf(wv);
  ws[WS_SQW + t] = sh_sqw[t];
  __syncthreads();
  for (int s = 64; s > 0; s >>= 1) { if (t < s) red[t] += red[t + s]; __syncthreads(); }
  if (t == 0) { sc[0] = red[0]; ws[WS_WSUM] = red[0]; }
  __syncthreads();
  const float wsum = sc[0];

  // b_w[k][j] = sum_n b[k,j,n]*w[n] / wsum
  for (int idx = t; idx < 192; idx += 128) {
    int k = idx / 3, j = idx % 3;
    const float* row = bkp + k * 384 + j * 128;
    float s = 0.f;
    for (int n = 0; n < 128; n++) s += row[n] * sh_sqw[n] * sh_sqw[n];
    s /= wsum;
    sh_bw[idx] = s;
    ws[WS_BW + idx] = s;
  }
  __syncthreads();

  // bar_B[m=3n+j][k] = sqw[n]*(b[k,j,n]-b_w[k,j])
  for (int idx = t; idx < 24576; idx += 128) {
    int m = idx >> 6, k = idx & 63;
    int n = m / 3, j = m % 3;
    ws[WS_BARB + idx] = sh_sqw[n] * (bkp[k * 384 + j * 128 + n] - sh_bw[k * 3 + j]);
  }
  __syncthreads();

  // bar_H = 2*(B^T B + lam I); also fill augmented [H | I]
  for (int idx = t; idx < 4096; idx += 128) {
    int k = idx >> 6, k2 = idx & 63;
    float s = 0.f;
    for (int m = 0; m < 384; m++) s += ws[WS_BARB + m * 64 + k] * ws[WS_BARB + m * 64 + k2];
    float h = 2.f * (s + ((k == k2) ? LAM : 0.f));
    ws[WS_BARH + idx] = h;
    aug[k][k2] = h;
    aug[k][64 + k2] = (k == k2) ? 1.f : 0.f;
  }
  __syncthreads();

  // Gauss-Jordan (SPD -> no pivoting)
  for (int i = 0; i < 64; i++) {
    if (t < 64) fac[t] = aug[t][i];
    __syncthreads();
    float piv = fac[i];
    aug[i][t] = aug[i][t] / piv;
    __syncthreads();
    float rowi = aug[i][t];
    for (int r = 0; r < 64; r++)
      if (r != i) aug[r][t] -= fac[r] * rowi;
    __syncthreads();
  }

  // Htemp = rowsum(Hinv); s = sum; g = Htemp/s
  if (t < 64) {
    float s = 0.f;
    for (int k2 = 0; k2 < 64; k2++) s += aug[t][64 + k2];
    fac[t] = s;
  }
  __syncthreads();
  if (t == 0) {
    float s = 0.f;
    for (int k = 0; k < 64; k++) s += fac[k];
    sc[1] = s;
  }
  __syncthreads();
  const float stot = sc[1];
  if (t < 64) { sh_g[t] = fac[t] / stot; ws[WS_G0 + t] = sh_g[t]; }
  __syncthreads();

  // G = Hinv - Htemp Htemp^T / s  (into aug cols 0..63)
  for (int idx = t; idx < 4096; idx += 128) {
    int k = idx >> 6, k2 = idx & 63;
    aug[k][k2] = aug[k][64 + k2] - fac[k] * fac[k2] / stot;
  }
  __syncthreads();

  // A_mat[k][m] = sum_k2 G[k][k2]*barB[m][k2]
  for (int idx = t; idx < 24576; idx += 128) {
    int k = idx / 384, m = idx % 384;
    float s = 0.f;
    for (int k2 = 0; k2 < 64; k2++) s += aug[k][k2] * ws[WS_BARB + m * 64 + k2];
    ws[WS_AMAT + idx] = s;
  }
  __syncthreads();

  // htop[m] = sum_k barB[m][k]*g[k]
  for (int m = t; m < 384; m += 128) {
    float s = 0.f;
    for (int k = 0; k < 64; k++) s += ws[WS_BARB + m * 64 + k] * sh_g[k];
    sh_ht[m] = s;
    ws[WS_HT + m] = s;
  }
  __syncthreads();

  // u[k] = sum_m htop[m]*barB[m][k]
  if (t < 64) {
    float s = 0.f;
    for (int m = 0; m < 384; m++) s += sh_ht[m] * ws[WS_BARB + m * 64 + t];
    sh_u[t] = s;
  }
  __syncthreads();

  // mh = M^T h = 2 A^T u - htop + 2*sqrt(lam) A^T g
  const float twosql = 2.f * sqrtf(LAM);
  for (int m = t; m < 384; m += 128) {
    float s1 = 0.f, s2 = 0.f;
    for (int k = 0; k < 64; k++) {
      float a = ws[WS_AMAT + k * 384 + m];
      s1 += sh_u[k] * a;
      s2 += sh_g[k] * a;
    }
    ws[WS_MH + m] = 2.f * s1 - sh_ht[m] + twosql * s2;
  }
  if (t == 0) {
    float s = 0.f;
    for (int m = 0; m < 384; m++) s += sh_ht[m] * sh_ht[m];
    for (int k = 0; k < 64; k++) s += sh_g[k] * sh_g[k];
    ws[WS_Q00] = s;
  }
  __syncthreads();

  // BH_A[k][m] = sum_k2 barH[k][k2]*A[k2][m]
  for (int idx = t; idx < 24576; idx += 128) {
    int k = idx / 384, m = idx % 384;
    float s = 0.f;
    for (int k2 = 0; k2 < 64; k2++) s += ws[WS_BARH + k * 64 + k2] * ws[WS_AMAT + k2 * 384 + m];
    ws[WS_BHA + idx] = s;
  }
}

// ============================================================ prep2: S = M^T M
__global__ __launch_bounds__(256) void pace_prep2(float* __restrict__ ws) {
  int idx = blockIdx.x * 256 + threadIdx.x;
  if (idx >= 147456) return;
  int m = idx / 384, m2 = idx % 384;
  const float* A   = ws + WS_AMAT;
  const float* B   = ws + WS_BARB;
  const float* BHA = ws + WS_BHA;
  float s1 = 0.f, s2 = 0.f, s3 = 0.f;
  for (int k = 0; k < 64; k++) {
    s1 += A[k * 384 + m] * BHA[k * 384 + m2];
    s2 += B[m * 64 + k]  * A[k * 384 + m2];
    s3 += B[m2 * 64 + k] * A[k * 384 + m];
  }
  ws[WS_S + idx] = 2.f * s1 - 2.f * (s2 + s3) + ((m == m2) ? 1.f : 0.f);
}

// ============================================================ prep3: interleave S_pq^T
// Bil[pq][kp][n][e] = S_pq^T[2kp+e][n] = S[(3n+p)*384 + 3*(2kp+e) + q]
// flat: WS_SPQ + (((pq*64 + kp)*128 + n)*2 + e)
__global__ __launch_bounds__(256) void pace_prep3(float* __restrict__ ws) {
  int idx = blockIdx.x * 256 + threadIdx.x;
  if (idx >= 147456) return;
  int e  = idx & 1;
  int n  = (idx >> 1) & 127;
  int kp = (idx >> 8) & 63;
  int pq = idx >> 14;               // 0..8
  int p = pq / 3, q = pq % 3;
  int k = 2 * kp + e;
  ws[WS_SPQ + idx] = ws[WS_S + (3 * n + p) * 384 + 3 * k + q];
}

// ============================================================ mainQ: bar_y, q0, Q22 via WMMA
__global__ __launch_bounds__(128) void pace_mainQ(const float* __restrict__ y,
                                                  float* __restrict__ ws) {
  __shared__ float X[48][130];
  __shared__ float H[48][130];
  __shared__ float yw[48];
  const int t = threadIdx.x;
  const int b0 = blockIdx.x * 16;
  const float* sqw = ws + WS_SQW;
  const float* mh  = ws + WS_MH;
  const float wsum = ws[WS_WSUM];
  const float Q00  = ws[WS_Q00];

  // y_w for 48 rows
  if (t < 48) {
    int bl = t / 3, j = t % 3;
    const float* yr = y + (size_t)(b0 + bl) * 384 + j * 128;
    float s = 0.f;
    for (int n = 0; n < 128; n++) s += yr[n] * sqw[n] * sqw[n];
    s /= wsum;
    yw[t] = s;
    ws[WS_YW + (b0 + bl) * 3 + j] = s;
  }
  __syncthreads();

  // X = bar_y tile (also spill to ws for later kernels)
  for (int idx = t; idx < 6144; idx += 128) {
    int r = idx >> 7, n = idx & 127;
    int bl = r / 3, j = r % 3;
    float v = sqw[n] * (y[(size_t)(b0 + bl) * 384 + j * 128 + n] - yw[r]);
    X[r][n] = v;
    ws[WS_BARY + (size_t)(b0 + bl) * 384 + j * 128 + n] = v;
  }
  __syncthreads();

  // q0[b,c] = sum_n mh[3n + c/3] * bar_y[b, c%3, n]; fill Q row0/col0 + Q00
  for (int idx = t; idx < 144; idx += 128) {
    int bl = idx / 9, c = idx % 9;
    int p = c / 3, jc = c % 3;
    const float* xr = &X[3 * bl + jc][0];
    float s = 0.f;
    for (int n = 0; n < 128; n++) s += mh[3 * n + p] * xr[n];
    int gb = b0 + bl;
    float* Qb = ws + WS_QBUF + (size_t)gb * 100;
    Qb[0 * 10 + (1 + c)] = s;
    Qb[(1 + c) * 10 + 0] = s;
    if (c == 0) Qb[0] = Q00;
  }

  const int lane = t & 31, wv = t >> 5;
  const int khalf = (lane >> 4) * 2;   // 0 or 2; kk = k0 + khalf is always even

  for (int pq = 0; pq < 9; pq++) {
    const int p = pq / 3, q = pq % 3;
    __syncthreads();   // protect H from previous iteration's readers

    // H = X @ S_pq^T : M=48, N=128, K=128 via V_WMMA_F32_16X16X4_F32
    const float* Bil = ws + WS_SPQ + (size_t)pq * 16384;
    for (int tile = wv; tile < 24; tile += 4) {
      int mt = tile / 8, nt = tile % 8;
      int lrow = mt * 16 + (lane & 15);            // A row (M)
      int lcol = nt * 16 + (lane & 15);            // B col (N)
      const float* Bp = Bil + lcol * 2;            // + (kk/2)*256 per k-pair
      v8f acc = {};
      for (int k0 = 0; k0 < 128; k0 += 4) {
        int kk = k0 + khalf;
        v2f a, b;
        a.x = X[lrow][kk];
        a.y = X[lrow][kk + 1];
        b = *(const v2f*)(Bp + (kk >> 1) * 256);   // coalesced b64, const offsets
        acc = __builtin_amdgcn_wmma_f32_16x16x4_f32(
            false, a, false, b, (short)0, acc, false, false);
      }
      int rbase = mt * 16 + ((lane >> 4) ? 8 : 0);
      int cc = nt * 16 + (lane & 15);
      #pragma unroll
      for (int v = 0; v < 8; v++) H[rbase + v][cc] = acc[v];
    }
    __syncthreads();

    // T_pq[b][j][j'] = sum_n X[3b+j][n]*H[3b+j'][n]  -> Q22 entry
    for (int idx = t; idx < 144; idx += 128) {
      int bl = idx / 9, jj = idx % 9;
      int j = jj / 3, j2 = jj % 3;
      const float* xr = &X[3 * bl + j][0];
      const float* hr = &H[3 * bl + j2][0];
      float s = 0.f;
      for (int n = 0; n < 128; n++) s += xr[n] * hr[n];
      int gb = b0 + bl;
      ws[WS_QBUF + (size_t)gb * 100 + (1 + 3 * p + j) * 10 + (1 + 3 * q + j2)] = s;
    }
  }
}

// ============================================================ eigk: Jacobi 10x10, min-eigvec -> R
__global__ __launch_bounds__(64) void pace_eig(const float* __restrict__ ws,
                                               float* __restrict__ out) {
  int b = blockIdx.x * 64 + threadIdx.x;
  if (b >= BATCH) return;
  const float* Qs = ws + WS_QBUF + (size_t)b * 100;
  float A[10][10], V[10][10];
  for (int i = 0; i < 10; i++)
    for (int j = 0; j < 10; j++) {
      A[i][j] = 0.5f * (Qs[i * 10 + j] + Qs[j * 10 + i]);
      V[i][j] = (i == j) ? 1.f : 0.f;
    }
  for (int sweep = 0; sweep < 15; sweep++) {
    for (int p = 0; p < 9; p++) {
      for (int q = p + 1; q < 10; q++) {
        float apq = A[p][q];
        if (fabsf(apq) < 1e-14f) continue;
        float theta = (A[q][q] - A[p][p]) / (2.f * apq);
        float tt = ((theta >= 0.f) ? 1.f : -1.f) / (fabsf(theta) + sqrtf(theta * theta + 1.f));
        float c = 1.f / sqrtf(tt * tt + 1.f);
        float s = tt * c;
        for (int i = 0; i < 10; i++) {
          float aip = A[i][p], aiq = A[i][q];
          A[i][p] = c * aip - s * aiq;
          A[i][q] = s * aip + c * aiq;
        }
        for (int i = 0; i < 10; i++) {
          float api = A[p][i], aqi = A[q][i];
          A[p][i] = c * api - s * aqi;
          A[q][i] = s * api + c * aqi;
        }
        for (int i = 0; i < 10; i++) {
          float vip = V[i][p], viq = V[i][q];
          V[i][p] = c * vip - s * viq;
          V[i][q] = s * vip + c * viq;
        }
      }
    }
  }
  int best = 0;
  float bv = A[0][0];
  for (int i = 1; i < 10; i++) if (A[i][i] < bv) { bv = A[i][i]; best = i; }
  float inv = 1.f / V[0][best];
  for (int i = 0; i < 9; i++)
    out[(size_t)b * 9 + i] = V[1 + i][best] * inv;   // R row-major
}

// ============================================================ ctk: c and t
__global__ __launch_bounds__(64) void pace_ct(const float* __restrict__ ws,
                                              float* __restrict__ out) {
  __shared__ float Rt[384];
  __shared__ float Rs[9];
  __shared__ float cs[64];
  __shared__ float bwc[3];
  const int b = blockIdx.x, t = threadIdx.x;
  const float* bar = ws + WS_BARY + (size_t)b * 384;
  if (t < 9) Rs[t] = out[(size_t)b * 9 + t];
  __syncthreads();
  for (int n = t; n < 128; n += 64) {
    float y0 = bar[n], y1 = bar[128 + n], y2 = bar[256 + n];
    #pragma unroll
    for (int j = 0; j < 3; j++)
      Rt[3 * n + j] = Rs[j] * y0 + Rs[3 + j] * y1 + Rs[6 + j] * y2;  // sum_i R[i][j]*bar_y[i][n]
  }
  __syncthreads();
  {
    const float* A = ws + WS_AMAT + (size_t)t * 384;
    float acc = 0.f;
    for (int m = 0; m < 384; m++) acc += A[m] * Rt[m];
    float ck = 2.f * acc + ws[WS_G0 + t];
    cs[t] = ck;
    out[BATCH * 12 + (size_t)b * 64 + t] = ck;           // c at offset 98304
  }
  __syncthreads();
  if (t < 3) {
    float s = 0.f;
    for (int k = 0; k < 64; k++) s += ws[WS_BW + k * 3 + t] * cs[k];
    bwc[t] = s;
  }
  __syncthreads();
  if (t < 3) {
    float tv = ws[WS_YW + (size_t)b * 3 + t]
             - (Rs[t * 3] * bwc[0] + Rs[t * 3 + 1] * bwc[1] + Rs[t * 3 + 2] * bwc[2]);
    out[BATCH * 9 + (size_t)b * 3 + t] = tv;             // t at offset 73728
  }
}

// ============================================================ launch
extern "C" void kernel_launch(void* const* d_in, const int* in_sizes, int n_in,
                              void* d_out, int out_size, void* d_ws, size_t ws_size,
                              hipStream_t stream) {
  const float* y  = (const float*)d_in[0];   // (8192,3,128)
  const float* w  = (const float*)d_in[1];   // (128,1)
  const float* mk = (const float*)d_in[2];   // (64,3,128)
  float* out = (float*)d_out;
  float* ws  = (float*)d_ws;

  pace_prep1<<<1, 128, 0, stream>>>(w, mk, ws);
  pace_prep2<<<(147456 + 255) / 256, 256, 0, stream>>>(ws);
  pace_prep3<<<(147456 + 255) / 256, 256, 0, stream>>>(ws);
  pace_mainQ<<<BATCH / 16, 128, 0, stream>>>(y, ws);
  pace_eig<<<BATCH / 64, 64, 0, stream>>>(ws, out);
  pace_ct<<<BATCH, 64, 0, stream>>>(ws, out);
}